// ReservoirCell_24232205484530
// MI455X (gfx1250) — compile-verified
//
#include <hip/hip_runtime.h>
#include <cstdint>
#include <math.h>

#define UNITS   2048
#define D_IN    96
#define BATCH   64
#define T_STEPS 512
#define C_REC   10
#define NNZ_IN  (96 * 10)          // 960
#define M_ROWS  (BATCH * T_STEPS)  // 32768

typedef __attribute__((ext_vector_type(2))) float v2f;
typedef __attribute__((ext_vector_type(8))) float v8f;

// ---------------- Kernel 1a: zero dense input kernel (in d_ws) ----------------
__global__ void zero_kd(float* __restrict__ kd) {
    int i = blockIdx.x * blockDim.x + threadIdx.x;
    if (i < D_IN * UNITS) kd[i] = 0.0f;
}

// ---------------- Kernel 1b: scatter COO nnz into dense kernel ----------------
__global__ void scatter_kd(float* __restrict__ kd, const float* __restrict__ vals,
                           const int* __restrict__ rows, const int* __restrict__ cols) {
    int i = blockIdx.x * blockDim.x + threadIdx.x;
    if (i < NNZ_IN) kd[rows[i] * UNITS + cols[i]] = vals[i];
}

// ---------------- Kernel 2: pre[m,n] = bias[n] + X[m,:] @ Kd  (WMMA f32 16x16x4)
// M = 32768, N = 2048, K = 96. One 16x16 C tile per wave; 8 waves/block along N.
// Writes straight into d_out (recurrent pass updates it in place afterwards).
__global__ __launch_bounds__(256)
void input_gemm(const float* __restrict__ X, const float* __restrict__ Kd,
                const float* __restrict__ bias, float* __restrict__ C) {
    const int wave = threadIdx.x >> 5;
    const int lane = threadIdx.x & 31;
    const int half = lane >> 4;          // selects K pair (A) / K rows (B), C row half
    const int l    = lane & 15;          // M index (A) / N index (B, C)
    const int ngrp  = blockIdx.x & 15;   // 16 groups of 8 N-tiles = 128 N-tiles
    const int mtile = blockIdx.x >> 4;   // 2048 M-tiles
    const int n0 = (ngrp * 8 + wave) * 16;
    const int m0 = mtile * 16;

    const float bv = bias[n0 + l];
    v8f acc = { bv, bv, bv, bv, bv, bv, bv, bv };   // C[m,n] init = bias[n]

    const float* xrow = X + (size_t)(m0 + l) * D_IN;
    #pragma unroll
    for (int k = 0; k < D_IN; k += 4) {
        v2f a, b;
        // A 16x4: lanes 0-15 -> K=k,k+1 ; lanes 16-31 -> K=k+2,k+3 (row M = l)
        a.x = xrow[k + 2 * half + 0];
        a.y = xrow[k + 2 * half + 1];
        // B 4x16: lanes 0-15 -> K rows k,k+1 ; lanes 16-31 -> K rows k+2,k+3 (col N = n0+l)
        b.x = Kd[(size_t)(k + 2 * half + 0) * UNITS + n0 + l];
        b.y = Kd[(size_t)(k + 2 * half + 1) * UNITS + n0 + l];
        acc = __builtin_amdgcn_wmma_f32_16x16x4_f32(
                  false, a, false, b, (short)0, acc, false, false);
    }
    // C layout: VGPR i -> row m0 + i + 8*half, column n0 + l
    float* crow = C + (size_t)(m0 + 8 * half) * UNITS + n0 + l;
    #pragma unroll
    for (int i = 0; i < 8; ++i)
        crow[(size_t)i * UNITS] = acc[i];
}

// ---------------- Kernel 3: sequential reservoir scan, one block per batch ----
// h double-buffered in LDS; rec_idx/rec_vals pinned in VGPRs; next-step `pre`
// row prefetched from global while gathers/tanh of the current step execute.
__global__ __launch_bounds__(512, 1)
void reservoir_scan(const float* __restrict__ state0, const int* __restrict__ rec_idx,
                    const float* __restrict__ rec_vals, float* __restrict__ out) {
    __shared__ float hbuf[2][UNITS];     // 16 KB of the 320 KB WGP LDS
    const int b   = blockIdx.x;
    const int tid = threadIdx.x;

    int   idx[4][C_REC];
    float val[4][C_REC];
    #pragma unroll
    for (int j = 0; j < 4; ++j) {
        const int u = tid + j * 512;
        #pragma unroll
        for (int c = 0; c < C_REC; ++c) {
            idx[j][c] = rec_idx[u * C_REC + c];
            val[j][c] = rec_vals[u * C_REC + c];
        }
        hbuf[0][u] = state0[(size_t)b * UNITS + u];
    }
    __syncthreads();

    float* obase = out + (size_t)b * T_STEPS * UNITS;
    float pr[4];
    #pragma unroll
    for (int j = 0; j < 4; ++j) pr[j] = obase[tid + j * 512];   // pre for t=0

    int p = 0;
    for (int t = 0; t < T_STEPS; ++t) {
        float* orow = obase + (size_t)t * UNITS;
        float prn[4] = {0.f, 0.f, 0.f, 0.f};
        if (t + 1 < T_STEPS) {           // prefetch next step's input_part row
            const float* nrow = orow + UNITS;
            #pragma unroll
            for (int j = 0; j < 4; ++j) prn[j] = nrow[tid + j * 512];
        }
        const float* cur = hbuf[p];
        float*       nxt = hbuf[p ^ 1];
        #pragma unroll
        for (int j = 0; j < 4; ++j) {
            const int u = tid + j * 512;
            float s = pr[j];
            #pragma unroll
            for (int c = 0; c < C_REC; ++c)
                s += cur[idx[j][c]] * val[j][c];   // LDS gathers
            const float hn = tanhf(s);             // LEAKY == 1.0
            nxt[u]  = hn;
            orow[u] = hn;                          // overwrite pre with output
        }
        __syncthreads();
        #pragma unroll
        for (int j = 0; j < 4; ++j) pr[j] = prn[j];
        p ^= 1;
    }
}

extern "C" void kernel_launch(void* const* d_in, const int* in_sizes, int n_in,
                              void* d_out, int out_size, void* d_ws, size_t ws_size,
                              hipStream_t stream) {
    const float* X      = (const float*)d_in[0];   // [64,512,96]
    const float* state0 = (const float*)d_in[1];   // [64,2048]
    const float* kvals  = (const float*)d_in[2];   // [960]
    const float* rvals  = (const float*)d_in[3];   // [2048,10]
    const float* bias   = (const float*)d_in[4];   // [2048]
    const int*   krows  = (const int*)d_in[5];     // [960]
    const int*   kcols  = (const int*)d_in[6];     // [960]
    const int*   ridx   = (const int*)d_in[7];     // [2048,10]
    float* out = (float*)d_out;                    // [64,512,2048] f32
    float* kd  = (float*)d_ws;                     // dense input kernel, 768 KB

    zero_kd   <<<(D_IN * UNITS + 255) / 256, 256, 0, stream>>>(kd);
    scatter_kd<<<(NNZ_IN + 255) / 256,       256, 0, stream>>>(kd, kvals, krows, kcols);
    // 2048 M-tiles * 16 N-groups = 32768 blocks, 8 waves each -> one 16x16 tile/wave
    input_gemm<<<32768, 256, 0, stream>>>(X, kd, bias, out);
    reservoir_scan<<<BATCH, 512, 0, stream>>>(state0, ridx, rvals, out);
}